// DeepSeekV2Attention_40750649704538
// MI455X (gfx1250) — compile-verified
//
#include <hip/hip_runtime.h>
#include <hip/hip_bf16.h>

typedef __attribute__((ext_vector_type(2))) float v2f;
typedef __attribute__((ext_vector_type(8))) float v8f;
typedef __attribute__((ext_vector_type(4))) int   v4i;

#define NHEAD   32
#define DHEAD   128
#define HIDDEN  5120
#define QSZ     (NHEAD * DHEAD)      // 4096
#define QKVO    4672                 // QSZ + 512 + 64
#define BSZ     64
#define SMAX    8192
#define CHUNK   1024
#define MAXCH   (SMAX / CHUNK)       // 8
#define SCALE_F 0.08838834764831845f // 128^-0.5
#define NEGINF  (-1e30f)

// ---- optional gfx1250 async global->LDS path (guarded; falls back) --------
#if defined(__has_builtin)
#if __has_builtin(__builtin_amdgcn_global_load_async_to_lds_b128)
#define HAVE_ASYNC_LDS 1
#endif
#endif

#ifdef HAVE_ASYNC_LDS
typedef __attribute__((address_space(1))) v4i* as1_v4i;
typedef __attribute__((address_space(3))) v4i* as3_v4i;

static __device__ __forceinline__ void async_ld128(const float* gsrc,
                                                   float* ldst) {
  // Global generic pointer value == AS1 address; low 32 bits of a generic
  // LDS pointer == LDS offset (AS3). Integer round-trip performs the casts.
  __builtin_amdgcn_global_load_async_to_lds_b128(
      (as1_v4i)(unsigned long long)gsrc,
      (as3_v4i)(unsigned int)(unsigned long long)ldst, 0, 0);
}
static __device__ __forceinline__ void wait_async0() {
#if __has_builtin(__builtin_amdgcn_s_wait_asynccnt)
  __builtin_amdgcn_s_wait_asynccnt(0);
#else
  asm volatile("s_wait_asynccnt 0" ::: "memory");
#endif
}
static __device__ __forceinline__ void fence_ds0() {
  asm volatile("s_wait_dscnt 0" ::: "memory");
}
#endif

static __device__ __forceinline__ v8f wmma4(v2f a, v2f b, v8f c) {
  // V_WMMA_F32_16X16X4_F32 : D = A(16x4) * B(4x16) + C(16x16), all f32
  return __builtin_amdgcn_wmma_f32_16x16x4_f32(false, a, false, b, (short)0, c,
                                               false, false);
}

// ---------------------------------------------------------------------------
// C(64 x N) = A(64 x K) * W(N x K)^T         (row-major A, W, C)
// grid.x = N/16 ; block = 128 threads (4 waves, one 16-row M-tile per wave)
// ---------------------------------------------------------------------------
__global__ __launch_bounds__(128) void gemm64_nt(const float* __restrict__ A,
                                                 const float* __restrict__ W,
                                                 float* __restrict__ C,
                                                 int N, int K) {
  __shared__ float sW[16 * 68]; // 16 rows of W, K-chunk 64, pad 4
  __shared__ float sA[64 * 68]; // 64 rows of A, K-chunk 64, pad 4

  const int tid  = threadIdx.x;
  const int lane = tid & 31;
  const int w    = tid >> 5;
  const int llo  = lane & 15;
  const int kb   = (lane >> 4) << 1; // K sub-offset {0,2} per lane-half
  const long n0  = (long)blockIdx.x * 16;

  v8f acc = {0.f, 0.f, 0.f, 0.f, 0.f, 0.f, 0.f, 0.f};

  for (int kc = 0; kc < K; kc += 64) {
#ifdef HAVE_ASYNC_LDS
    // ---- async staging: global -> LDS without VGPR round-trip ----
#pragma unroll
    for (int i = 0; i < 2; ++i) {
      int f4 = tid + 128 * i;
      int r = f4 >> 4, c = (f4 & 15) << 2;
      const float* src = W + (n0 + r) * (long)K + kc + c;
      if (kc + 64 < K) __builtin_prefetch(src + 64, 0, 1); // global_prefetch
      async_ld128(src, &sW[r * 68 + c]);
    }
#pragma unroll
    for (int i = 0; i < 8; ++i) {
      int f4 = tid + 128 * i;
      int r = f4 >> 4, c = (f4 & 15) << 2;
      async_ld128(A + r * (long)K + kc + c, &sA[r * 68 + c]);
    }
    wait_async0();
#else
    // ---- fallback: two-phase (batch loads, then stores) for high MLP ----
    float4 tw[2], ta[8];
#pragma unroll
    for (int i = 0; i < 2; ++i) {
      int f4 = tid + 128 * i;
      int r = f4 >> 4, c = (f4 & 15) << 2;
      const float* src = W + (n0 + r) * (long)K + kc + c;
      if (kc + 64 < K) __builtin_prefetch(src + 64, 0, 1); // global_prefetch
      tw[i] = *(const float4*)src;
    }
#pragma unroll
    for (int i = 0; i < 8; ++i) {
      int f4 = tid + 128 * i;
      int r = f4 >> 4, c = (f4 & 15) << 2;
      ta[i] = *(const float4*)(A + r * (long)K + kc + c);
    }
#pragma unroll
    for (int i = 0; i < 2; ++i) {
      int f4 = tid + 128 * i;
      int r = f4 >> 4, c = (f4 & 15) << 2;
      *(float4*)&sW[r * 68 + c] = tw[i];
    }
#pragma unroll
    for (int i = 0; i < 8; ++i) {
      int f4 = tid + 128 * i;
      int r = f4 >> 4, c = (f4 & 15) << 2;
      *(float4*)&sA[r * 68 + c] = ta[i];
    }
#endif
    __syncthreads();

    const int m = (w << 4) + llo;
#pragma unroll
    for (int kk = 0; kk < 16; ++kk) {
      v2f a  = *(const v2f*)&sA[m * 68 + kk * 4 + kb];
      v2f bb = *(const v2f*)&sW[llo * 68 + kk * 4 + kb];
      acc = wmma4(a, bb, acc);
    }
    __syncthreads();
  }

  // C/D layout: VGPR r holds row (r + 8*(lane>=16)), col = lane&15
  const long nc = n0 + llo;
  const int  rb = (w << 4) + ((lane >> 4) << 3);
#pragma unroll
  for (int r = 0; r < 8; ++r)
    C[(long)(rb + r) * N + nc] = acc[r];
}

// ---------------------------------------------------------------------------
// Stage one 16x128 K/V tile into wave-private LDS (stride 132, zero-padded
// rows beyond `end`, slot row L-1 sourced from kv_new).
// ---------------------------------------------------------------------------
static __device__ __forceinline__ void stage_kv_tile(
    float* sK, const float* cache, const float* kvnew, long b, int s0, int end,
    int L, int lane) {
#ifdef HAVE_ASYNC_LDS
#pragma unroll
  for (int r = 0; r < 16; ++r) {
    int s = s0 + r;
    float* dst = sK + r * 132 + (lane << 2);
    if (s < end) {
      const float* src =
          (s == L - 1) ? (kvnew + (lane << 2))
                       : (cache + (b * SMAX + s) * DHEAD + (lane << 2));
      async_ld128(src, dst);
    } else {
      float4 z = {0.f, 0.f, 0.f, 0.f};
      *(float4*)dst = z;
    }
  }
  wait_async0();
#else
#pragma unroll
  for (int rr = 0; rr < 16; rr += 4) {
    float4 tmp[4];
#pragma unroll
    for (int r = 0; r < 4; ++r) {
      int s = s0 + rr + r;
      float4 v = {0.f, 0.f, 0.f, 0.f};
      if (s < end) {
        const float* src =
            (s == L - 1) ? kvnew : (cache + (b * SMAX + s) * DHEAD);
        v = *(const float4*)(src + (lane << 2));
      }
      tmp[r] = v;
    }
#pragma unroll
    for (int r = 0; r < 4; ++r)
      *(float4*)(sK + (rr + r) * 132 + (lane << 2)) = tmp[r];
  }
#endif
}

// ---------------------------------------------------------------------------
// Flash-decoding chunk kernel.
// grid = (MAXCH, BSZ); block = 128 (4 waves). Wave w handles seq tiles
// start + 16*w, start + 16*w + 64, ...  (wave-private LDS staging).
// Writes per-chunk partials (m[32], l[32], acc[32x128]) to workspace.
// ---------------------------------------------------------------------------
__global__ __launch_bounds__(128) void attn_chunk(
    const float* __restrict__ qkv, const float* __restrict__ kcache,
    const float* __restrict__ vcache, const int* __restrict__ seq_lens,
    float* __restrict__ pm, float* __restrict__ pl, float* __restrict__ pacc) {
  __shared__ float sQ[32 * 132];     // Q (32 heads x 128), pad 4; reused as sAcc
  __shared__ float sKV[4][16 * 132]; // per-wave K/V tile (16 x 128), pad 4
  __shared__ float sP[4][16 * 18];   // per-wave P tile (16 x 16), pad 2
  __shared__ float sMS[4][32][2];    // per-wave row stats (m, l)

  const int b  = blockIdx.y;
  const int ch = blockIdx.x;
  const int L  = seq_lens[b];
  const int start = ch * CHUNK;
  if (start >= L) return;
  const int end = (start + CHUNK < L) ? (start + CHUNK) : L;

  const int tid = threadIdx.x, lane = tid & 31, w = tid >> 5;
  const int llo = lane & 15, lhi = lane >> 4, kb = lhi << 1;

  const float* qb    = qkv + (long)b * QKVO;
  const float* kvnew = qb + QSZ; // fresh K=V row for slot = L-1

  // stage Q (32 x 128) cooperatively
#pragma unroll
  for (int i = 0; i < 8; ++i) {
    int f4 = tid + 128 * i;
    int r = f4 >> 5, c = (f4 & 31) << 2;
    float4 v = *(const float4*)(qb + r * DHEAD + c);
    *(float4*)&sQ[r * 132 + c] = v;
  }
  __syncthreads();

  const v8f z8 = {0.f, 0.f, 0.f, 0.f, 0.f, 0.f, 0.f, 0.f};
  v8f acc0[8], acc1[8];
  float m0[8], m1[8], l0[8], l1[8];
#pragma unroll
  for (int r = 0; r < 8; ++r) {
    acc0[r] = z8; acc1[r] = z8;
    m0[r] = NEGINF; m1[r] = NEGINF; l0[r] = 0.f; l1[r] = 0.f;
  }

  float* sK  = sKV[w];
  float* sPw = sP[w];

  for (int s0 = start + 16 * w; s0 < end; s0 += 64) {
    // ---- stage K tile (wave-private) ----
    stage_kv_tile(sK, kcache, kvnew, b, s0, end, L, lane);

    // ---- S(32x16) = Q(32x128) * K_tile^T : 64 WMMAs ----
    v8f c0 = z8, c1 = z8;
#pragma unroll
    for (int kk = 0; kk < 32; ++kk) {
      v2f bf = *(const v2f*)&sK[llo * 132 + kk * 4 + kb]; // B[k=d][n=seq]
      v2f a0 = *(const v2f*)&sQ[llo * 132 + kk * 4 + kb];
      v2f a1 = *(const v2f*)&sQ[(16 + llo) * 132 + kk * 4 + kb];
      c0 = wmma4(a0, bf, c0);
      c1 = wmma4(a1, bf, c1);
    }

    // ---- online softmax (rows live in VGPR r, cols across 16-lane group) ---
    const bool vcol = (s0 + llo) < end;
    float sc0[8], sc1[8];
#pragma unroll
    for (int r = 0; r < 8; ++r) {
      float x0 = vcol ? c0[r] * SCALE_F : NEGINF;
      float x1 = vcol ? c1[r] * SCALE_F : NEGINF;
      float v0 = x0, v1 = x1;
#pragma unroll
      for (int msk = 1; msk < 16; msk <<= 1) {
        v0 = fmaxf(v0, __shfl_xor(v0, msk, 32));
        v1 = fmaxf(v1, __shfl_xor(v1, msk, 32));
      }
      float nm0 = fmaxf(m0[r], v0), nm1 = fmaxf(m1[r], v1);
      float p0 = __expf(x0 - nm0), p1 = __expf(x1 - nm1);
      c0[r] = p0; c1[r] = p1;
      float t0 = p0, t1 = p1;
#pragma unroll
      for (int msk = 1; msk < 16; msk <<= 1) {
        t0 += __shfl_xor(t0, msk, 32);
        t1 += __shfl_xor(t1, msk, 32);
      }
      sc0[r] = __expf(m0[r] - nm0);
      sc1[r] = __expf(m1[r] - nm1);
      l0[r] = l0[r] * sc0[r] + t0;
      l1[r] = l1[r] * sc1[r] + t1;
      m0[r] = nm0; m1[r] = nm1;
    }
#pragma unroll
    for (int nd = 0; nd < 8; ++nd)
#pragma unroll
      for (int r = 0; r < 8; ++r) {
        acc0[nd][r] *= sc0[r];
        acc1[nd][r] *= sc1[r];
      }

    // ---- stage V tile (overwrites K tile) ----
#ifdef HAVE_ASYNC_LDS
    // async-engine LDS writes are not ordered against our prior ds_loads of
    // the K tile by the compiler; fence DS reads before overwriting.
    fence_ds0();
#endif
    stage_kv_tile(sK, vcache, kvnew, b, s0, end, L, lane);

    // ---- PV: acc(16x128) += P(16x16) * V(16x128), per head tile ----
    // head tile 0: C-layout -> A-layout through padded LDS
#pragma unroll
    for (int r = 0; r < 8; ++r) sPw[(r + 8 * lhi) * 18 + llo] = c0[r];
#pragma unroll
    for (int k0 = 0; k0 < 16; k0 += 4) {
      v2f a = *(const v2f*)&sPw[llo * 18 + k0 + kb];
#pragma unroll
      for (int nd = 0; nd < 8; ++nd) {
        v2f bv;
        bv.x = sK[(k0 + kb) * 132 + (nd << 4) + llo];
        bv.y = sK[(k0 + kb + 1) * 132 + (nd << 4) + llo];
        acc0[nd] = wmma4(a, bv, acc0[nd]);
      }
    }
    // head tile 1
#pragma unroll
    for (int r = 0; r < 8; ++r) sPw[(r + 8 * lhi) * 18 + llo] = c1[r];
#pragma unroll
    for (int k0 = 0; k0 < 16; k0 += 4) {
      v2f a = *(const v2f*)&sPw[llo * 18 + k0 + kb];
#pragma unroll
      for (int nd = 0; nd < 8; ++nd) {
        v2f bv;
        bv.x = sK[(k0 + kb) * 132 + (nd << 4) + llo];
        bv.y = sK[(k0 + kb + 1) * 132 + (nd << 4) + llo];
        acc1[nd] = wmma4(a, bv, acc1[nd]);
      }
    }
  }

  // ---- combine the 4 waves' partial softmax states ----
  if (llo == 0) {
#pragma unroll
    for (int r = 0; r < 8; ++r) {
      sMS[w][r + 8 * lhi][0] = m0[r];      sMS[w][r + 8 * lhi][1] = l0[r];
      sMS[w][16 + r + 8 * lhi][0] = m1[r]; sMS[w][16 + r + 8 * lhi][1] = l1[r];
    }
  }
  __syncthreads();

  float csc0[8], csc1[8];
#pragma unroll
  for (int r = 0; r < 8; ++r) {
    int row0 = r + 8 * lhi, row1 = 16 + row0;
    float mm0 = NEGINF, mm1 = NEGINF;
    for (int ww = 0; ww < 4; ++ww) {
      mm0 = fmaxf(mm0, sMS[ww][row0][0]);
      mm1 = fmaxf(mm1, sMS[ww][row1][0]);
    }
    csc0[r] = __expf(m0[r] - mm0);
    csc1[r] = __expf(m1[r] - mm1);
  }

  float* sAcc = sQ; // reuse Q region (32x128 fits in 32x132)
  for (int it = 0; it < 4; ++it) {
    if (w == it) {
#pragma unroll
      for (int nd = 0; nd < 8; ++nd)
#pragma unroll
        for (int r = 0; r < 8; ++r) {
          int col = (nd << 4) + llo;
          int row0 = r + 8 * lhi, row1 = 16 + row0;
          float v0 = acc0[nd][r] * csc0[r];
          float v1 = acc1[nd][r] * csc1[r];
          if (it) { v0 += sAcc[row0 * 128 + col]; v1 += sAcc[row1 * 128 + col]; }
          sAcc[row0 * 128 + col] = v0;
          sAcc[row1 * 128 + col] = v1;
        }
    }
    __syncthreads();
  }

  const long pb = (long)b * MAXCH + ch;
  if (tid < 32) {
    float mm = NEGINF;
    for (int ww = 0; ww < 4; ++ww) mm = fmaxf(mm, sMS[ww][tid][0]);
    float ll = 0.f;
    for (int ww = 0; ww < 4; ++ww)
      ll += sMS[ww][tid][1] * __expf(sMS[ww][tid][0] - mm);
    pm[pb * 32 + tid] = mm;
    pl[pb * 32 + tid] = ll;
  }
  float* pa = pacc + pb * 4096;
  for (int i = tid; i < 4096; i += 128) pa[i] = sAcc[i];
}

// ---------------------------------------------------------------------------
// Combine chunk partials -> attn (64 x 32 x 128)
// ---------------------------------------------------------------------------
__global__ __launch_bounds__(256) void attn_reduce(
    const float* __restrict__ pm, const float* __restrict__ pl,
    const float* __restrict__ pacc, const int* __restrict__ seq_lens,
    float* __restrict__ attn) {
  const int b = blockIdx.x;
  const int L = seq_lens[b];
  const int nch = (L + CHUNK - 1) / CHUNK;
  for (int idx = threadIdx.x; idx < QSZ; idx += 256) {
    const int h = idx >> 7;
    float M = NEGINF;
    for (int c = 0; c < nch; ++c)
      M = fmaxf(M, pm[((long)b * MAXCH + c) * 32 + h]);
    float lsum = 0.f, asum = 0.f;
    for (int c = 0; c < nch; ++c) {
      float sc = __expf(pm[((long)b * MAXCH + c) * 32 + h] - M);
      lsum += pl[((long)b * MAXCH + c) * 32 + h] * sc;
      asum += pacc[((long)b * MAXCH + c) * 4096 + idx] * sc;
    }
    attn[(long)b * QSZ + idx] = asum / lsum;
  }
}

// ---------------------------------------------------------------------------
extern "C" void kernel_launch(void* const* d_in, const int* in_sizes, int n_in,
                              void* d_out, int out_size, void* d_ws,
                              size_t ws_size, hipStream_t stream) {
  (void)in_sizes; (void)n_in; (void)out_size; (void)ws_size;
  const float* hidden  = (const float*)d_in[0];
  // d_in[1] = positions (unused by reference math)
  const float* kvcache = (const float*)d_in[2];
  // d_in[3] = slot_mapping (== seq_lens - 1)
  const int*   seqlen  = (const int*)d_in[4];
  const float* w_qkv   = (const float*)d_in[5];
  const float* w_o     = (const float*)d_in[6];
  float*       out     = (float*)d_out;

  float* ws   = (float*)d_ws;
  float* qkv  = ws;                         // 64 * 4672
  float* attn = qkv + (long)BSZ * QKVO;     // 64 * 4096
  float* pm   = attn + (long)BSZ * QSZ;     // 64 * 8 * 32
  float* pl   = pm + (long)BSZ * MAXCH * 32;
  float* pacc = pl + (long)BSZ * MAXCH * 32; // 64 * 8 * 4096

  const float* kc = kvcache;
  const float* vc = kvcache + (long)BSZ * SMAX * DHEAD;

  // 1) QKV projection: qkv(64 x 4672) = hidden(64 x 5120) * w_qkv^T
  gemm64_nt<<<QKVO / 16, 128, 0, stream>>>(hidden, w_qkv, qkv, QKVO, HIDDEN);
  // 2) flash-decoding attention over chunks
  attn_chunk<<<dim3(MAXCH, BSZ), 128, 0, stream>>>(qkv, kc, vc, seqlen, pm, pl,
                                                   pacc);
  // 3) combine chunk partials
  attn_reduce<<<BSZ, 256, 0, stream>>>(pm, pl, pacc, seqlen, attn);
  // 4) output projection: out(64 x 5120) = attn(64 x 4096) * w_o^T
  gemm64_nt<<<HIDDEN / 16, 128, 0, stream>>>(attn, w_o, out, HIDDEN, QSZ);
}